// EulerMotion_RBF_69947837383262
// MI455X (gfx1250) — compile-verified
//
#include <hip/hip_runtime.h>
#include <math.h>

// ---------------------------------------------------------------------------
// RBF interpolation (scipy Rbf, gaussian, N-D, smooth=0) on gfx1250.
//   centers (4096,3) f32, vel (4096,3) f32, query (65536,3) f32 -> out (65536,3)
// Pipeline: prep (eps, |c|^2, rhs=vel) -> build A via WMMA -> blocked LU
// (panel / trsm / 32x32 WMMA gemm / rhs update) -> blocked back-substitution
// -> pack (cn,nodes) -> fused WMMA evaluation with TDM-staged LDS table.
// ---------------------------------------------------------------------------

#define NPTS 4096
#define MQ   65536
#define NB   32

typedef __attribute__((ext_vector_type(2))) float v2f;
typedef __attribute__((ext_vector_type(8))) float v8f;
typedef __attribute__((ext_vector_type(4))) unsigned int v4u;
typedef __attribute__((ext_vector_type(4))) int v4i;
typedef __attribute__((ext_vector_type(8))) int v8i;

__device__ __forceinline__ v8f wmma4(v2f a, v2f b, v8f c) {
  // D(16x16,f32) = A(16x4,f32) x B(4x16,f32) + C   -> v_wmma_f32_16x16x4_f32
  return __builtin_amdgcn_wmma_f32_16x16x4_f32(false, a, false, b, (short)0, c,
                                               false, false);
}

// ---------------------------------------------------------------------------
// K1: bbox reduction -> inv_eps2; cnorm; rhs = vel
// ---------------------------------------------------------------------------
__global__ __launch_bounds__(256) void rbf_prep(const float* __restrict__ centers,
                                                const float* __restrict__ vel,
                                                float* __restrict__ cnorm,
                                                float* __restrict__ rhs,
                                                float* __restrict__ scal) {
  __shared__ float smn[3][256];
  __shared__ float smx[3][256];
  int tid = threadIdx.x;
  float mn0 = 1e30f, mn1 = 1e30f, mn2 = 1e30f;
  float mx0 = -1e30f, mx1 = -1e30f, mx2 = -1e30f;
  for (int i = tid; i < NPTS; i += 256) {
    float x = centers[i * 3 + 0];
    float y = centers[i * 3 + 1];
    float z = centers[i * 3 + 2];
    cnorm[i] = x * x + y * y + z * z;
    mn0 = fminf(mn0, x); mx0 = fmaxf(mx0, x);
    mn1 = fminf(mn1, y); mx1 = fmaxf(mx1, y);
    mn2 = fminf(mn2, z); mx2 = fmaxf(mx2, z);
  }
  smn[0][tid] = mn0; smn[1][tid] = mn1; smn[2][tid] = mn2;
  smx[0][tid] = mx0; smx[1][tid] = mx1; smx[2][tid] = mx2;
  __syncthreads();
  for (int s = 128; s > 0; s >>= 1) {
    if (tid < s) {
      for (int a = 0; a < 3; a++) {
        smn[a][tid] = fminf(smn[a][tid], smn[a][tid + s]);
        smx[a][tid] = fmaxf(smx[a][tid], smx[a][tid + s]);
      }
    }
    __syncthreads();
  }
  if (tid == 0) {
    float e0 = smx[0][0] - smn[0][0];
    float e1 = smx[1][0] - smn[1][0];
    float e2 = smx[2][0] - smn[2][0];
    float eps = cbrtf(e0 * e1 * e2 / (float)NPTS);
    scal[0] = 1.0f / (eps * eps);
  }
  for (int i = tid; i < NPTS * 3; i += 256) rhs[i] = vel[i];
}

// ---------------------------------------------------------------------------
// K2: A[i][j] = exp(-(|ci|^2+|cj|^2-2 ci.cj)/eps^2), one 16x16 tile per wave.
// ---------------------------------------------------------------------------
__global__ __launch_bounds__(128) void rbf_build_a(const float* __restrict__ centers,
                                                   const float* __restrict__ cnorm,
                                                   const float* __restrict__ scal,
                                                   float* __restrict__ A) {
  int wave = (blockIdx.x * blockDim.x + threadIdx.x) >> 5;
  int lane = threadIdx.x & 31;
  int li = lane & 15;
  bool hi = lane >= 16;
  int ti = wave >> 8;       // row tile (256 tiles)
  int tj = wave & 255;      // col tile
  float ie2 = scal[0];

  const float* ca = centers + (size_t)(ti * 16 + li) * 3;  // A rows = centers[ti]
  v2f af; af.x = hi ? ca[2] : ca[0]; af.y = hi ? 0.f : ca[1];
  const float* cb = centers + (size_t)(tj * 16 + li) * 3;  // B cols = centers[tj]
  v2f bf; bf.x = hi ? cb[2] : cb[0]; bf.y = hi ? 0.f : cb[1];

  v8f d = {};
  d = wmma4(af, bf, d);

  float cnj = cnorm[tj * 16 + li];
  int ibase = ti * 16 + (hi ? 8 : 0);
#pragma unroll
  for (int r = 0; r < 8; r++) {
    float cni = cnorm[ibase + r];
    float d2 = fmaxf(cni + cnj - 2.0f * d[r], 0.0f);
    A[(size_t)(ibase + r) * NPTS + tj * 16 + li] = __expf(-d2 * ie2);
  }
}

// ---------------------------------------------------------------------------
// LU (no pivot; A is SPD): panel factor, cols [k0,k0+NB), rows [k0,N)
// ---------------------------------------------------------------------------
__global__ __launch_bounds__(1024) void lu_panel(float* __restrict__ A, int k0) {
  __shared__ float piv[NB];
  __shared__ float sinv;
  int tid = threadIdx.x;
  for (int j = 0; j < NB; j++) {
    int k = k0 + j;
    if (tid == 0) sinv = 1.0f / A[(size_t)k * NPTS + k];
    if (tid < NB) piv[tid] = (tid > j) ? A[(size_t)k * NPTS + k0 + tid] : 0.0f;
    __syncthreads();
    float inv = sinv;
    for (int i = k + 1 + tid; i < NPTS; i += 1024) {
      size_t row = (size_t)i * NPTS;
      float m = A[row + k] * inv;
      A[row + k] = m;
      for (int jj = j + 1; jj < NB; jj++) A[row + k0 + jj] -= m * piv[jj];
    }
    __syncthreads();
  }
}

// ---------------------------------------------------------------------------
// U12 = L11^{-1} A12 (unit-lower trsm), rhs treated as 3 trailing columns.
// One column per thread, L11 staged in LDS, column held in registers.
// ---------------------------------------------------------------------------
__global__ __launch_bounds__(256) void lu_trsm(float* __restrict__ A,
                                               float* __restrict__ rhs, int k0) {
  __shared__ float L[NB * NB];
  int tid = threadIdx.x;
  for (int idx = tid; idx < NB * NB; idx += 256)
    L[idx] = A[(size_t)(k0 + idx / NB) * NPTS + k0 + (idx % NB)];
  __syncthreads();

  int cid = blockIdx.x * 256 + tid;
  int ncols = (NPTS - k0 - NB) + 3;
  if (cid >= ncols) return;
  int c = k0 + NB + cid;
  bool isr = c >= NPTS;
  int jr = c - NPTS;

  float col[NB];
#pragma unroll
  for (int r = 0; r < NB; r++)
    col[r] = isr ? rhs[(k0 + r) * 3 + jr] : A[(size_t)(k0 + r) * NPTS + c];
#pragma unroll
  for (int r = 1; r < NB; r++)
#pragma unroll
    for (int s = 0; s < r; s++) col[r] -= L[r * NB + s] * col[s];
#pragma unroll
  for (int r = 0; r < NB; r++) {
    if (isr) rhs[(k0 + r) * 3 + jr] = col[r];
    else     A[(size_t)(k0 + r) * NPTS + c] = col[r];
  }
}

// ---------------------------------------------------------------------------
// Trailing update A22 -= L21 * U12, WMMA f32 16x16x4, 32x32 tile per wave
// (4 accumulators -> 4 WMMAs per 8 per-lane loads per K=4 step).
// ---------------------------------------------------------------------------
__global__ __launch_bounds__(128) void lu_gemm(float* __restrict__ A, int k0) {
  int wave = (blockIdx.x * blockDim.x + threadIdx.x) >> 5;
  int lane = threadIdx.x & 31;
  int li = lane & 15;
  bool hi = lane >= 16;

  int R = NPTS - k0 - NB;   // remaining rows/cols, multiple of 32
  int T = R >> 5;           // 32x32 tiles per dim
  if (wave >= T * T) return;
  int tr = wave / T;
  int tc = wave % T;
  int i0 = k0 + NB + tr * 32;
  int c0 = k0 + NB + tc * 32;

  v8f acc00 = {}, acc01 = {}, acc10 = {}, acc11 = {};
  const float* Ar0 = A + (size_t)(i0 + li) * NPTS;
  const float* Ar1 = A + (size_t)(i0 + 16 + li) * NPTS;
#pragma unroll
  for (int s8 = 0; s8 < NB / 4; s8++) {
    int s = k0 + s8 * 4 + (hi ? 2 : 0);
    v2f a0; a0.x = Ar0[s]; a0.y = Ar0[s + 1];
    v2f a1; a1.x = Ar1[s]; a1.y = Ar1[s + 1];
    const float* Bs0 = A + (size_t)s * NPTS;
    const float* Bs1 = A + (size_t)(s + 1) * NPTS;
    v2f b0; b0.x = Bs0[c0 + li];      b0.y = Bs1[c0 + li];
    v2f b1; b1.x = Bs0[c0 + 16 + li]; b1.y = Bs1[c0 + 16 + li];
    acc00 = wmma4(a0, b0, acc00);
    acc01 = wmma4(a0, b1, acc01);
    acc10 = wmma4(a1, b0, acc10);
    acc11 = wmma4(a1, b1, acc11);
  }
#pragma unroll
  for (int r = 0; r < 8; r++) {
    int ra = i0 + r + (hi ? 8 : 0);
    int rb = ra + 16;
    A[(size_t)ra * NPTS + c0 + li]      -= acc00[r];
    A[(size_t)ra * NPTS + c0 + 16 + li] -= acc01[r];
    A[(size_t)rb * NPTS + c0 + li]      -= acc10[r];
    A[(size_t)rb * NPTS + c0 + 16 + li] -= acc11[r];
  }
}

// ---------------------------------------------------------------------------
// rhs trailing update: rhs[i] -= L21[i,:] . y  for rows below the panel.
// ---------------------------------------------------------------------------
__global__ __launch_bounds__(256) void lu_rhs_update(const float* __restrict__ A,
                                                     float* __restrict__ rhs,
                                                     int k0) {
  __shared__ float yb[NB * 3];
  int tid = threadIdx.x;
  if (tid < NB * 3) yb[tid] = rhs[k0 * 3 + tid];
  __syncthreads();
  int i = k0 + NB + blockIdx.x * 256 + tid;
  if (i >= NPTS) return;
  float s0 = rhs[i * 3 + 0], s1 = rhs[i * 3 + 1], s2 = rhs[i * 3 + 2];
  for (int t = 0; t < NB; t++) {
    float a = A[(size_t)i * NPTS + k0 + t];
    s0 -= a * yb[t * 3 + 0];
    s1 -= a * yb[t * 3 + 1];
    s2 -= a * yb[t * 3 + 2];
  }
  rhs[i * 3 + 0] = s0; rhs[i * 3 + 1] = s1; rhs[i * 3 + 2] = s2;
}

// ---------------------------------------------------------------------------
// Back-substitution (U x = y), blocked: small triangular solve + rank-NB update
// ---------------------------------------------------------------------------
__global__ void bs_small(const float* __restrict__ A, float* __restrict__ rhs,
                         int k0) {
  int j = threadIdx.x;
  if (j >= 3) return;
  for (int r = NB - 1; r >= 0; r--) {
    float s = rhs[(k0 + r) * 3 + j];
    for (int t = r + 1; t < NB; t++)
      s -= A[(size_t)(k0 + r) * NPTS + k0 + t] * rhs[(k0 + t) * 3 + j];
    rhs[(k0 + r) * 3 + j] = s / A[(size_t)(k0 + r) * NPTS + k0 + r];
  }
}

__global__ __launch_bounds__(256) void bs_update(const float* __restrict__ A,
                                                 float* __restrict__ rhs, int k0) {
  __shared__ float xb[NB * 3];
  int tid = threadIdx.x;
  if (tid < NB * 3) xb[tid] = rhs[k0 * 3 + tid];
  __syncthreads();
  int i = blockIdx.x * 256 + tid;
  if (i >= k0) return;
  float s0 = rhs[i * 3 + 0], s1 = rhs[i * 3 + 1], s2 = rhs[i * 3 + 2];
  for (int t = 0; t < NB; t++) {
    float a = A[(size_t)i * NPTS + k0 + t];
    s0 -= a * xb[t * 3 + 0];
    s1 -= a * xb[t * 3 + 1];
    s2 -= a * xb[t * 3 + 2];
  }
  rhs[i * 3 + 0] = s0; rhs[i * 3 + 1] = s1; rhs[i * 3 + 2] = s2;
}

// ---------------------------------------------------------------------------
// Pack (cnorm, nodes) into a float4 table for the evaluation pass.
// ---------------------------------------------------------------------------
__global__ __launch_bounds__(256) void pack_nodes(const float* __restrict__ cnorm,
                                                  const float* __restrict__ rhs,
                                                  float4* __restrict__ packed) {
  int i = blockIdx.x * 256 + threadIdx.x;
  if (i < NPTS)
    packed[i] = make_float4(cnorm[i], rhs[i * 3 + 0], rhs[i * 3 + 1], rhs[i * 3 + 2]);
}

// ---------------------------------------------------------------------------
// Fused evaluation: out[q] = sum_c exp(-d2(q,c)/eps^2) * nodes[c]
// Wave owns 16 queries (A frag, loaded once). (cn,nx,ny,nz) table (64KB) is
// staged into LDS once per block via the Tensor Data Mover; per center tile:
// one WMMA for the 16x16 dots, one ds_load_b128 per lane, per-lane phi +
// within-lane FMAs, shuffle reduction at the end.
// ---------------------------------------------------------------------------
__global__ __launch_bounds__(256) void rbf_eval(const float* __restrict__ query,
                                                const float* __restrict__ centers,
                                                const float4* __restrict__ packed,
                                                const float* __restrict__ scal,
                                                float* __restrict__ out) {
  __shared__ float4 spack[NPTS];  // 64 KB: (cnorm, nx, ny, nz) per center
  int tid = threadIdx.x;

#if defined(__has_builtin) && __has_builtin(__builtin_amdgcn_tensor_load_to_lds) && \
    __has_builtin(__builtin_amdgcn_s_wait_tensorcnt)
  if (tid < 32) {  // wave 0 issues one 64KB TDM transfer (EXEC ignored by TDM)
    unsigned long long ga = (unsigned long long)(uintptr_t)packed;
    unsigned int lds = (unsigned int)(uintptr_t)&spack[0];
    // D# group 0: count=1 | lds_addr | global_addr[56:0] | type=2
    v4u g0;
    g0.x = 1u;
    g0.y = lds;
    g0.z = (unsigned int)ga;
    g0.w = (unsigned int)((ga >> 32) & 0x1FFFFFFu) | (2u << 30);
    // D# group 1: 1-D tile, data_size=4B, tensor_dim0=tile_dim0=16384 dwords
    v8i g1;
    g1[0] = (int)(2u << 16);                        // data_size = 4 bytes
    g1[1] = (int)((16384u & 0xFFFFu) << 16);        // tensor_dim0 lo
    g1[2] = (int)((16384u >> 16) | (1u << 16));     // tensor_dim0 hi | tensor_dim1 lo = 1
    g1[3] = (int)(16384u << 16);                    // tensor_dim1 hi=0 | tile_dim0
    g1[4] = 0;                                      // tile_dim1 = 0 (1-D), tile_dim2 = 0
    g1[5] = 16384;                                  // tensor_dim0_stride lo
    g1[6] = 0;
    g1[7] = 0;
    // 6-arg form (clang-23 / therock-10.0 headers): two unused descriptor
    // groups (int32x4) + trailing int32x8 group, then cache-policy imm.
    __builtin_amdgcn_tensor_load_to_lds(g0, g1, (v4i)0, (v4i)0, (v8i)0, 0);
    __builtin_amdgcn_s_wait_tensorcnt(0);
  }
#else
  for (int i = tid; i < NPTS; i += 256) spack[i] = packed[i];
#endif
  __syncthreads();

  int wave = (blockIdx.x * blockDim.x + tid) >> 5;
  int lane = tid & 31;
  int li = lane & 15;
  bool hi = lane >= 16;
  int qbase = wave * 16;
  float ie2 = scal[0];

  const float* qp = query + (size_t)(qbase + li) * 3;
  v2f af; af.x = hi ? qp[2] : qp[0]; af.y = hi ? 0.f : qp[1];

  float qn[8];
#pragma unroll
  for (int r = 0; r < 8; r++) {
    const float* q = query + (size_t)(qbase + r + (hi ? 8 : 0)) * 3;
    qn[r] = q[0] * q[0] + q[1] * q[1] + q[2] * q[2];
  }

  float acc[8][3] = {};
  for (int t = 0; t < NPTS / 16; t++) {
    int cb = t * 16 + li;
    const float* cp = centers + (size_t)cb * 3;
    __builtin_prefetch(cp + 48, 0, 0);  // next center tile -> global_prefetch_b8
    v2f bf; bf.x = hi ? cp[2] : cp[0]; bf.y = hi ? 0.f : cp[1];
    v8f d = {};
    d = wmma4(af, bf, d);
    float4 cd = spack[cb];              // (cn, nx, ny, nz) via ds_load_b128
#pragma unroll
    for (int r = 0; r < 8; r++) {
      float d2 = fmaxf(qn[r] + cd.x - 2.0f * d[r], 0.0f);
      float phi = __expf(-d2 * ie2);
      acc[r][0] += phi * cd.y;
      acc[r][1] += phi * cd.z;
      acc[r][2] += phi * cd.w;
    }
  }
#pragma unroll
  for (int r = 0; r < 8; r++)
#pragma unroll
    for (int j = 0; j < 3; j++) {
      float v = acc[r][j];
      v += __shfl_xor(v, 1);
      v += __shfl_xor(v, 2);
      v += __shfl_xor(v, 4);
      v += __shfl_xor(v, 8);
      acc[r][j] = v;
    }
  if (li == 0) {
#pragma unroll
    for (int r = 0; r < 8; r++) {
      int q = qbase + r + (hi ? 8 : 0);
      out[q * 3 + 0] = acc[r][0];
      out[q * 3 + 1] = acc[r][1];
      out[q * 3 + 2] = acc[r][2];
    }
  }
}

// ---------------------------------------------------------------------------
extern "C" void kernel_launch(void* const* d_in, const int* in_sizes, int n_in,
                              void* d_out, int out_size, void* d_ws, size_t ws_size,
                              hipStream_t stream) {
  const float* centers = (const float*)d_in[0];  // (4096,3)
  const float* vel     = (const float*)d_in[1];  // (4096,3)
  const float* query   = (const float*)d_in[2];  // (65536,3)
  float* out = (float*)d_out;                    // (65536,3)

  // workspace layout: A (N*N) | packed (N float4) | rhs (N*3) | cnorm (N) | scal
  float*  A      = (float*)d_ws;
  float4* packed = (float4*)(A + (size_t)NPTS * NPTS);
  float*  rhs    = (float*)(packed + NPTS);
  float*  cn     = rhs + NPTS * 3;
  float*  scal   = cn + NPTS;

  rbf_prep<<<1, 256, 0, stream>>>(centers, vel, cn, rhs, scal);

  // build A: 256x256 tiles, 1 wave each, 4 waves/block
  rbf_build_a<<<(256 * 256) / 4, 128, 0, stream>>>(centers, cn, scal, A);

  // blocked LU with rhs carried along
  for (int k0 = 0; k0 < NPTS; k0 += NB) {
    lu_panel<<<1, 1024, 0, stream>>>(A, k0);
    int ncols = (NPTS - k0 - NB) + 3;
    lu_trsm<<<(ncols + 255) / 256, 256, 0, stream>>>(A, rhs, k0);
    int R = NPTS - k0 - NB;
    if (R > 0) {
      int tiles = (R >> 5) * (R >> 5);
      lu_gemm<<<(tiles + 3) / 4, 128, 0, stream>>>(A, k0);
      lu_rhs_update<<<(R + 255) / 256, 256, 0, stream>>>(A, rhs, k0);
    }
  }

  // back-substitution (nodes end up in rhs)
  for (int k0 = NPTS - NB; k0 >= 0; k0 -= NB) {
    bs_small<<<1, 32, 0, stream>>>(A, rhs, k0);
    if (k0 > 0) bs_update<<<(k0 + 255) / 256, 256, 0, stream>>>(A, rhs, k0);
  }

  pack_nodes<<<NPTS / 256, 256, 0, stream>>>(cn, rhs, packed);

  // fused evaluation: 4096 waves (16 queries each), 8 waves/block
  rbf_eval<<<(MQ / 16) / 8, 256, 0, stream>>>(query, centers, packed, scal, out);
}